// RGCN_317827579998
// MI455X (gfx1250) — compile-verified
//
#include <hip/hip_runtime.h>

typedef float v2f __attribute__((ext_vector_type(2)));
typedef float v8f __attribute__((ext_vector_type(8)));

#define DFEAT 128
#define PSTR  144   // LDS pair-row stride in float2 units: 2*144 dwords % 64 banks = 32 -> halves disjoint

// ---------------- zero fill (float4 granularity, grid-stride) ----------------
__global__ void k_zero4(float4* __restrict__ p, long long n4) {
  long long i = (long long)blockIdx.x * blockDim.x + threadIdx.x;
  long long stride = (long long)gridDim.x * blockDim.x;
  float4 z; z.x = z.y = z.z = z.w = 0.f;
  for (; i < n4; i += stride) p[i] = z;
}

// ---------------- degree counting via hardware f32 atomics ----------------
__global__ void k_degree(const int* __restrict__ src, const int* __restrict__ dst,
                         float* __restrict__ outdeg, float* __restrict__ indeg, int n_edges) {
  int i = blockIdx.x * blockDim.x + threadIdx.x;
  int stride = gridDim.x * blockDim.x;
  for (int e = i; e < n_edges; e += stride) {
    unsafeAtomicAdd(&outdeg[src[e]], 1.f);
    unsafeAtomicAdd(&indeg[dst[e]],  1.f);
  }
}

__global__ void k_norm(float* __restrict__ outdeg, float* __restrict__ indeg, int n) {
  int i = blockIdx.x * blockDim.x + threadIdx.x;
  if (i < n) {
    outdeg[i] = rsqrtf(fmaxf(outdeg[i], 1.f));
    indeg[i]  = rsqrtf(fmaxf(indeg[i],  1.f));
  }
}

// H[m,:] = act(IN[m,:]) * out_norm[m] @ W, one 16-row strip per wave via V_WMMA_F32_16X16X4_F32.
// MODE 0: act(v) = v                              (layer 1, IN = x)
// MODE 1: act(v) = relu(v*in_norm[m] + bias[k])   (layers 2/3, IN = previous aggregation)
template <int MODE>
__global__ void __launch_bounds__(256) k_gemm_wmma(
    const float* __restrict__ IN, const float* __restrict__ Wg,
    const float* __restrict__ out_norm, const float* __restrict__ in_norm,
    const float* __restrict__ bias, float* __restrict__ H, int n_nodes)
{
  // W staged K-pair-interleaved: Ws[lp*PSTR + n] = {W[k0+2lp][n], W[k0+2lp+1][n]}
  __shared__ v2f Ws[32 * PSTR];  // 36 KB: one 64-K-row chunk of W
  const int tid  = threadIdx.x;
  const int wave = tid >> 5;
  const int lane = tid & 31;
  const int half = lane >> 4;     // which 16-lane half of the wave
  const int ln   = lane & 15;

  const int row_base = (blockIdx.x * 8 + wave) * 16;
  const int m  = row_base + ln;                       // A-matrix row owned by this lane
  const int mc = m < n_nodes ? m : (n_nodes - 1);     // clamped: loads always valid
  const float on  = (m < n_nodes) ? out_norm[mc] : 0.f;  // 0 kills invalid rows' contribution
  const float inr = MODE ? in_norm[mc] : 0.f;
  const float* __restrict__ inrow = IN + (size_t)mc * DFEAT;

  v8f acc[8] = {};  // 8 tiles of 16x16 -> 16 x 128 output strip

  for (int kc = 0; kc < 2; ++kc) {              // stage W rows [kc*64, kc*64+64) in LDS
    __syncthreads();
    for (int i = tid; i < 32 * DFEAT; i += 256) {  // 4096 float2 entries
      int lp = i >> 7, c = i & (DFEAT - 1);
      int k0 = kc * 64 + 2 * lp;
      v2f w;
      w.x = Wg[(k0 + 0) * DFEAT + c];
      w.y = Wg[(k0 + 1) * DFEAT + c];
      Ws[lp * PSTR + c] = w;
    }
    __syncthreads();

    for (int kb = 0; kb < 16; ++kb) {
      // A fragment: lane holds A[m][kg], A[m][kg+1]  (K = 2*half + vgpr per ISA layout)
      const int kg = kc * 64 + kb * 4 + half * 2;
      v2f a = *(const v2f*)(inrow + kg);  // 8B-aligned, always in-bounds (clamped row)
      if (MODE) {
        a.x = fmaxf(fmaf(a.x, inr, bias[kg]),     0.f);
        a.y = fmaxf(fmaf(a.y, inr, bias[kg + 1]), 0.f);
      }
      a.x *= on; a.y *= on;

      const int lp = kb * 2 + half;  // LDS pair-row of this lane's B fragment
#pragma unroll
      for (int t = 0; t < 8; ++t) {
        // single ds_load_b64 -> aligned VGPR pair {W[krow][n], W[krow+1][n]}
        v2f b = Ws[lp * PSTR + t * 16 + ln];
        acc[t] = __builtin_amdgcn_wmma_f32_16x16x4_f32(
            false, a, false, b, (short)0, acc[t], false, false);
      }
    }
  }

  // D layout: vgpr r of lane -> element (M = r + 8*half, N = ln) of each tile
#pragma unroll
  for (int t = 0; t < 8; ++t) {
    const int n0 = t * 16;
#pragma unroll
    for (int r = 0; r < 8; ++r) {
      const int mr = row_base + r + 8 * half;
      if (mr < n_nodes) H[(size_t)mr * DFEAT + n0 + ln] = acc[t][r];
    }
  }
}

// one wave per edge iteration: 32 lanes x float4 covers the 128-wide feature row,
// scatter-add with no-return global_atomic_add_f32 (h/agg are 25.6 MB -> L2-resident)
__global__ void k_scatter(const float* __restrict__ H, const int* __restrict__ src,
                          const int* __restrict__ dst, float* __restrict__ AGG, int n_edges)
{
  const int gtid = blockIdx.x * blockDim.x + threadIdx.x;
  const int wid  = gtid >> 5;
  const int lane = threadIdx.x & 31;
  const int nw   = (gridDim.x * blockDim.x) >> 5;
  for (int e = wid; e < n_edges; e += nw) {
    const int s = src[e], d = dst[e];
    const float4 v = *(const float4*)(H + (size_t)s * DFEAT + lane * 4);
    float* p = AGG + (size_t)d * DFEAT + lane * 4;
    unsafeAtomicAdd(p + 0, v.x);
    unsafeAtomicAdd(p + 1, v.y);
    unsafeAtomicAdd(p + 2, v.z);
    unsafeAtomicAdd(p + 3, v.w);
  }
}

__global__ void k_finalize(const float* __restrict__ agg, const float* __restrict__ in_norm,
                           const float* __restrict__ bias, float* __restrict__ out, long long n)
{
  long long i = (long long)blockIdx.x * blockDim.x + threadIdx.x;
  long long stride = (long long)gridDim.x * blockDim.x;
  for (; i < n; i += stride) {
    int row = (int)(i >> 7), col = (int)(i & (DFEAT - 1));
    out[i] = fmaf(agg[i], in_norm[row], bias[col]);
  }
}

extern "C" void kernel_launch(void* const* d_in, const int* in_sizes, int n_in,
                              void* d_out, int out_size, void* d_ws, size_t ws_size,
                              hipStream_t stream)
{
  const float* x   = (const float*)d_in[0];
  const int*   src = (const int*)d_in[1];
  const int*   dst = (const int*)d_in[2];
  const float* W1  = (const float*)d_in[3];
  const float* b1  = (const float*)d_in[4];
  const float* W2  = (const float*)d_in[5];
  const float* b2  = (const float*)d_in[6];
  const float* W3  = (const float*)d_in[7];
  const float* b3  = (const float*)d_in[8];
  float* out = (float*)d_out;

  const int n_nodes = in_sizes[0] / DFEAT;
  const int n_edges = in_sizes[1];
  const long long nd = (long long)n_nodes * DFEAT;

  // workspace: [out_norm N][in_norm N][h N*D][aggA N*D][aggB N*D]  (~77 MB for N=50000)
  float* ws   = (float*)d_ws;
  float* outn = ws;
  float* inn  = ws + n_nodes;
  float* h    = ws + 2 * (size_t)n_nodes;
  float* aggA = h + nd;
  float* aggB = aggA + nd;

  const int gemm_blocks = (n_nodes + 127) / 128;  // 8 waves/block * 16 rows/wave

  // degrees + norms (recomputed every call; ws is not re-poisoned but we re-zero)
  k_zero4<<<256, 256, 0, stream>>>((float4*)outn, (2LL * n_nodes + 3) / 4);
  k_degree<<<1024, 256, 0, stream>>>(src, dst, outn, inn, n_edges);
  k_norm<<<(n_nodes + 255) / 256, 256, 0, stream>>>(outn, inn, n_nodes);

  // layer 1
  k_gemm_wmma<0><<<gemm_blocks, 256, 0, stream>>>(x, W1, outn, inn, b1, h, n_nodes);
  k_zero4<<<2048, 256, 0, stream>>>((float4*)aggA, nd / 4);
  k_scatter<<<2048, 256, 0, stream>>>(h, src, dst, aggA, n_edges);

  // layer 2 (relu(aggA*in_norm + b1) fused into A-fragment load)
  k_gemm_wmma<1><<<gemm_blocks, 256, 0, stream>>>(aggA, W2, outn, inn, b1, h, n_nodes);
  k_zero4<<<2048, 256, 0, stream>>>((float4*)aggB, nd / 4);
  k_scatter<<<2048, 256, 0, stream>>>(h, src, dst, aggB, n_edges);

  // layer 3
  k_gemm_wmma<1><<<gemm_blocks, 256, 0, stream>>>(aggB, W3, outn, inn, b2, h, n_nodes);
  k_zero4<<<2048, 256, 0, stream>>>((float4*)aggA, nd / 4);
  k_scatter<<<2048, 256, 0, stream>>>(h, src, dst, aggA, n_edges);

  k_finalize<<<2048, 256, 0, stream>>>(aggA, inn, b3, out, nd);
}